// GAT_52656299049562
// MI455X (gfx1250) — compile-verified
//
#include <hip/hip_runtime.h>
#include <hip/hip_bf16.h>

#define NNODES 30000
#define NEDGES 240000

typedef float v2f __attribute__((ext_vector_type(2)));
typedef float v8f __attribute__((ext_vector_type(8)));

// ---------------------------------------------------------------------------
// WMMA f32 GEMM:  C[M,Nout] = A[M,K] * W[Nout,K]^T (+ bias[Nout])
// One wave32 computes a 16x(16*NT) tile of C via V_WMMA_F32_16X16X4_F32,
// reusing the A fragment across NT column tiles (NT=4 when Nout%64==0).
// K-loop is software-pipelined two k-steps deep (K must be a multiple of 8,
// true for all call sites: 128/256/512): all 2*(1+NT) fragment loads for the
// next k-pair issue before the 2*NT WMMAs that consume the current pair, so
// VMEM latency is fully covered by the matrix pipe.
// A-frag (16x4 f32, 2 VGPR): lane l -> row l&15, K pair 2*(l>>4)
// B-frag = W^T (4x16): mirrored layout -> float2 from W row (l&15)
// C/D  (16x16 f32, 8 VGPR): VGPR v -> row v + 8*(l>>4), col l&15
// ---------------------------------------------------------------------------
template <int NT>
__global__ void gat_gemm_wmma_f32(const float* __restrict__ A,
                                  const float* __restrict__ W,
                                  const float* __restrict__ bias,
                                  float* __restrict__ C,
                                  int M, int Nout, int K) {
    const int lane = threadIdx.x & 31;
    const int wave = blockIdx.x * (blockDim.x >> 5) + (threadIdx.x >> 5);
    const int tiles_m = M >> 4;
    const int tiles_n = Nout / (16 * NT);
    if (wave >= tiles_m * tiles_n) return;
    const int tm = wave % tiles_m;     // consecutive waves share the W tiles (L2 reuse)
    const int tg = wave / tiles_m;
    const int half = lane >> 4;        // 0 -> K {0,1}, 1 -> K {2,3}
    const int r    = lane & 15;        // A row within tile / B (=C) column within tile

    const float* __restrict__ arow = A + (size_t)(tm * 16 + r) * (size_t)K + 2 * half;
    const float* wrow[NT];
#pragma unroll
    for (int t = 0; t < NT; ++t)
        wrow[t] = W + (size_t)((tg * NT + t) * 16 + r) * (size_t)K + 2 * half;

    v8f acc[NT];
#pragma unroll
    for (int t = 0; t < NT; ++t) acc[t] = (v8f){0.f,0.f,0.f,0.f,0.f,0.f,0.f,0.f};

    // prologue: fragments for k = 0 and k = 4
    v2f a0 = *(const v2f*)(arow);
    v2f a1 = *(const v2f*)(arow + 4);
    v2f b0[NT], b1[NT];
#pragma unroll
    for (int t = 0; t < NT; ++t) b0[t] = *(const v2f*)(wrow[t]);
#pragma unroll
    for (int t = 0; t < NT; ++t) b1[t] = *(const v2f*)(wrow[t] + 4);

    for (int k = 8; k < K; k += 8) {
        // issue all loads for the next k-pair (independent of current WMMAs)
        const v2f an0 = *(const v2f*)(arow + k);
        const v2f an1 = *(const v2f*)(arow + k + 4);
        v2f bn0[NT], bn1[NT];
#pragma unroll
        for (int t = 0; t < NT; ++t) bn0[t] = *(const v2f*)(wrow[t] + k);
#pragma unroll
        for (int t = 0; t < NT; ++t) bn1[t] = *(const v2f*)(wrow[t] + k + 4);

        // consume current pair on the matrix pipe (2*NT back-to-back WMMAs)
#pragma unroll
        for (int t = 0; t < NT; ++t)
            acc[t] = __builtin_amdgcn_wmma_f32_16x16x4_f32(false, a0, false, b0[t],
                                                           (short)0, acc[t], false, false);
#pragma unroll
        for (int t = 0; t < NT; ++t)
            acc[t] = __builtin_amdgcn_wmma_f32_16x16x4_f32(false, a1, false, b1[t],
                                                           (short)0, acc[t], false, false);
        a0 = an0; a1 = an1;
#pragma unroll
        for (int t = 0; t < NT; ++t) { b0[t] = bn0[t]; b1[t] = bn1[t]; }
    }
    // epilogue: last k-pair
#pragma unroll
    for (int t = 0; t < NT; ++t)
        acc[t] = __builtin_amdgcn_wmma_f32_16x16x4_f32(false, a0, false, b0[t],
                                                       (short)0, acc[t], false, false);
#pragma unroll
    for (int t = 0; t < NT; ++t)
        acc[t] = __builtin_amdgcn_wmma_f32_16x16x4_f32(false, a1, false, b1[t],
                                                       (short)0, acc[t], false, false);

#pragma unroll
    for (int t = 0; t < NT; ++t) {
        const int col = (tg * NT + t) * 16 + r;
        const float bv = bias ? bias[col] : 0.f;
#pragma unroll
        for (int v = 0; v < 8; ++v) {
            const int row = tm * 16 + v + 8 * half;
            C[(size_t)row * (size_t)Nout + col] = acc[t][v] + bv;
        }
    }
}

// ---------------------------------------------------------------------------
// Attention scalars: el[n,h] = fs[n,h,:]·al[h,:], er likewise
// ---------------------------------------------------------------------------
__global__ void gat_node_attn(const float* __restrict__ fs,
                              const float* __restrict__ al,
                              const float* __restrict__ ar,
                              float* __restrict__ el, float* __restrict__ er,
                              int N, int H, int Fo) {
    const int i = blockIdx.x * blockDim.x + threadIdx.x;
    if (i >= N * H) return;
    const int n = i / H, h = i % H;
    const float* f = fs + (size_t)n * (size_t)(H * Fo) + h * Fo;
    const float* a = al + h * Fo;
    const float* b = ar + h * Fo;
    float sl = 0.f, sr = 0.f;
    for (int d = 0; d < Fo; ++d) { float v = f[d]; sl += v * a[d]; sr += v * b[d]; }
    el[i] = sl; er[i] = sr;
}

// order-preserving float<->uint encoding for atomicMax on f32
__device__ __forceinline__ unsigned f2ord(float f) {
    unsigned u = __float_as_uint(f);
    return (u & 0x80000000u) ? ~u : (u | 0x80000000u);
}
__device__ __forceinline__ float ord2f(unsigned u) {
    return (u & 0x80000000u) ? __uint_as_float(u & 0x7FFFFFFFu) : __uint_as_float(~u);
}
#define ORD_NEG_INF 0x007FFFFFu   // f2ord(-inf)

__device__ __forceinline__ float lrelu(float v) { return v > 0.f ? v : 0.2f * v; }

__global__ void gat_zero_f32(float* __restrict__ p, int n) {
    const int i = blockIdx.x * blockDim.x + threadIdx.x;
    if (i < n) p[i] = 0.f;
}
__global__ void gat_init_ms(unsigned* __restrict__ m, float* __restrict__ s, int n) {
    const int i = blockIdx.x * blockDim.x + threadIdx.x;
    if (i < n) { m[i] = ORD_NEG_INF; s[i] = 0.f; }
}

__global__ void gat_edge_max(const int* __restrict__ src, const int* __restrict__ dst,
                             const float* __restrict__ el, const float* __restrict__ er,
                             unsigned* __restrict__ m, int E, int H) {
    const int i = blockIdx.x * blockDim.x + threadIdx.x;
    if (i >= E * H) return;
    const int e = i / H, h = i % H;
    const float v = lrelu(el[src[e] * H + h] + er[dst[e] * H + h]);
    atomicMax(&m[dst[e] * H + h], f2ord(v));
}

__global__ void gat_edge_sum(const int* __restrict__ src, const int* __restrict__ dst,
                             const float* __restrict__ el, const float* __restrict__ er,
                             const unsigned* __restrict__ m, float* __restrict__ s,
                             int E, int H) {
    const int i = blockIdx.x * blockDim.x + threadIdx.x;
    if (i >= E * H) return;
    const int e = i / H, h = i % H;
    const int d = dst[e];
    const float v = lrelu(el[src[e] * H + h] + er[d * H + h]);
    atomicAdd(&s[d * H + h], expf(v - ord2f(m[d * H + h])));
}

__global__ void gat_edge_coef(const int* __restrict__ src, const int* __restrict__ dst,
                              const float* __restrict__ el, const float* __restrict__ er,
                              const unsigned* __restrict__ m, const float* __restrict__ s,
                              float* __restrict__ alpha, int E, int H) {
    const int i = blockIdx.x * blockDim.x + threadIdx.x;
    if (i >= E * H) return;
    const int e = i / H, h = i % H;
    const int d = dst[e];
    const float v = lrelu(el[src[e] * H + h] + er[d * H + h]);
    const float w = expf(v - ord2f(m[d * H + h]));
    alpha[i] = w / s[d * H + h];
}

// acc[dst, h, :] += alpha[e, h] * fs[src, h, :]; one thread per float4 chunk
__global__ void gat_edge_aggr(const int* __restrict__ src, const int* __restrict__ dst,
                              const float* __restrict__ alpha,
                              const float* __restrict__ fs, float* __restrict__ acc,
                              int E, int H, int Fo) {
    const int Q  = Fo >> 2;          // float4 chunks per head (Fo is 64 or 16)
    const int HQ = H * Q;
    const int total = E * HQ;
    const int i = blockIdx.x * blockDim.x + threadIdx.x;
    if (i >= total) return;
    const int e   = i / HQ;
    const int rem = i - e * HQ;
    const int h   = rem / Q;
    const int off = rem << 2;        // h*Fo + q*4 (flat within H*Fo)
    const int HF  = H * Fo;
    const float  a = alpha[e * H + h];
    const float4 f = *(const float4*)(fs + (size_t)src[e] * HF + off);
    float* p = acc + (size_t)dst[e] * HF + off;
    atomicAdd(p + 0, a * f.x);
    atomicAdd(p + 1, a * f.y);
    atomicAdd(p + 2, a * f.z);
    atomicAdd(p + 3, a * f.w);
}

// out[i] = act(acc[i] + bias[i % C] + res[i]), act = ELU or identity
__global__ void gat_finalize(const float* __restrict__ acc, const float* __restrict__ res,
                             const float* __restrict__ bias, float* __restrict__ out,
                             int total, int C, int do_elu) {
    const int i = blockIdx.x * blockDim.x + threadIdx.x;
    if (i >= total) return;
    float v = acc[i] + bias[i % C];
    if (res) v += res[i];
    if (do_elu) v = v > 0.f ? v : (expf(v) - 1.f);
    out[i] = v;
}

// ---------------------------------------------------------------------------
extern "C" void kernel_launch(void* const* d_in, const int* in_sizes, int n_in,
                              void* d_out, int out_size, void* d_ws, size_t ws_size,
                              hipStream_t stream) {
    (void)in_sizes; (void)n_in; (void)out_size; (void)ws_size;
    const float* feat0 = (const float*)d_in[0];
    const float* feat1 = (const float*)d_in[1];
    const float* feat2 = (const float*)d_in[2];
    const int*   src   = (const int*)d_in[3];
    const int*   dst   = (const int*)d_in[4];
    const float* fcW0  = (const float*)d_in[5];
    const float* fcb0  = (const float*)d_in[6];
    const float* fcW1  = (const float*)d_in[7];
    const float* fcb1  = (const float*)d_in[8];
    const float* fcW2  = (const float*)d_in[9];
    const float* fcb2  = (const float*)d_in[10];
    const float* W0  = (const float*)d_in[11];
    const float* al0 = (const float*)d_in[12];
    const float* ar0 = (const float*)d_in[13];
    const float* b0  = (const float*)d_in[14];
    const float* W1  = (const float*)d_in[15];
    const float* al1 = (const float*)d_in[16];
    const float* ar1 = (const float*)d_in[17];
    const float* b1  = (const float*)d_in[18];
    const float* W2  = (const float*)d_in[19];
    const float* al2 = (const float*)d_in[20];
    const float* ar2 = (const float*)d_in[21];
    const float* b2  = (const float*)d_in[22];
    const float* resW2 = (const float*)d_in[23];
    float* out = (float*)d_out;

    const int N = NNODES, E = NEDGES;

    // workspace carve-up (floats)
    float* ws = (float*)d_ws;
    size_t o = 0;
    float* H0 = ws + o; o += (size_t)N * 256;   // concat projected inputs
    float* FS = ws + o; o += (size_t)N * 512;   // per-layer fs (layer2: fs2 + res2)
    float* HA = ws + o; o += (size_t)N * 512;   // acc -> h1 (in place); later layer2 acc
    float* HB = ws + o; o += (size_t)N * 512;   // acc -> h2 (in place)
    float* EL = ws + o; o += (size_t)N * 8;
    float* ER = ws + o; o += (size_t)N * 8;
    unsigned* Menc = (unsigned*)(ws + o); o += (size_t)N * 8;
    float* S  = ws + o; o += (size_t)N * 8;
    float* AL = ws + o; o += (size_t)E * 8;     // alpha per (edge, head)

    const int TB = 256;
    auto gblk = [](int n, int tb) { return (n + tb - 1) / tb; };
    auto gemm = [&](const float* A, const float* Wm, const float* bias, float* C,
                    int M, int Nout, int K) {
        if ((Nout & 63) == 0) {
            const int tiles = (M >> 4) * (Nout >> 6);
            gat_gemm_wmma_f32<4><<<gblk(tiles, 8), 256, 0, stream>>>(A, Wm, bias, C, M, Nout, K);
        } else {
            const int tiles = (M >> 4) * (Nout >> 4);
            gat_gemm_wmma_f32<1><<<gblk(tiles, 8), 256, 0, stream>>>(A, Wm, bias, C, M, Nout, K);
        }
    };

    // ---- input projections: H0 = concat(feat_i @ fcW_i^T + fcb_i) ----
    gemm(feat0, fcW0, fcb0, H0,              10000, 256, 512);
    gemm(feat1, fcW1, fcb1, H0 + 10000*256,  10000, 256, 256);
    gemm(feat2, fcW2, fcb2, H0 + 20000*256,  10000, 256, 128);

    // ---- GAT layer 0: Fin=256, H=8, Fo=64, ELU, no residual ----
    {
        const int H = 8, Fo = 64, HF = 512;
        gemm(H0, W0, nullptr, FS, N, HF, 256);
        gat_node_attn<<<gblk(N*H, TB), TB, 0, stream>>>(FS, al0, ar0, EL, ER, N, H, Fo);
        gat_zero_f32<<<gblk(N*HF, TB), TB, 0, stream>>>(HA, N*HF);
        gat_init_ms<<<gblk(N*H, TB), TB, 0, stream>>>(Menc, S, N*H);
        gat_edge_max<<<gblk(E*H, TB), TB, 0, stream>>>(src, dst, EL, ER, Menc, E, H);
        gat_edge_sum<<<gblk(E*H, TB), TB, 0, stream>>>(src, dst, EL, ER, Menc, S, E, H);
        gat_edge_coef<<<gblk(E*H, TB), TB, 0, stream>>>(src, dst, EL, ER, Menc, S, AL, E, H);
        gat_edge_aggr<<<gblk(E*H*(Fo>>2), TB), TB, 0, stream>>>(src, dst, AL, FS, HA, E, H, Fo);
        gat_finalize<<<gblk(N*HF, TB), TB, 0, stream>>>(HA, nullptr, b0, HA, N*HF, HF, 1);
    }

    // ---- GAT layer 1: Fin=512, H=8, Fo=64, ELU, identity residual ----
    {
        const int H = 8, Fo = 64, HF = 512;
        gemm(HA, W1, nullptr, FS, N, HF, 512);
        gat_node_attn<<<gblk(N*H, TB), TB, 0, stream>>>(FS, al1, ar1, EL, ER, N, H, Fo);
        gat_zero_f32<<<gblk(N*HF, TB), TB, 0, stream>>>(HB, N*HF);
        gat_init_ms<<<gblk(N*H, TB), TB, 0, stream>>>(Menc, S, N*H);
        gat_edge_max<<<gblk(E*H, TB), TB, 0, stream>>>(src, dst, EL, ER, Menc, E, H);
        gat_edge_sum<<<gblk(E*H, TB), TB, 0, stream>>>(src, dst, EL, ER, Menc, S, E, H);
        gat_edge_coef<<<gblk(E*H, TB), TB, 0, stream>>>(src, dst, EL, ER, Menc, S, AL, E, H);
        gat_edge_aggr<<<gblk(E*H*(Fo>>2), TB), TB, 0, stream>>>(src, dst, AL, FS, HB, E, H, Fo);
        gat_finalize<<<gblk(N*HF, TB), TB, 0, stream>>>(HB, HA, b1, HB, N*HF, HF, 1);
    }

    // ---- GAT layer 2: Fin=512, H=1, Fo=16, no act, projected residual ----
    {
        const int H = 1, Fo = 16, HF = 16;
        float* FS2  = FS;                   // [N,16]
        float* RES2 = FS + (size_t)N * 16;  // [N,16]
        gemm(HB, W2,    nullptr, FS2,  N, HF, 512);
        gemm(HB, resW2, nullptr, RES2, N, HF, 512);
        gat_node_attn<<<gblk(N*H, TB), TB, 0, stream>>>(FS2, al2, ar2, EL, ER, N, H, Fo);
        gat_zero_f32<<<gblk(N*HF, TB), TB, 0, stream>>>(HA, N*HF);
        gat_init_ms<<<gblk(N*H, TB), TB, 0, stream>>>(Menc, S, N*H);
        gat_edge_max<<<gblk(E*H, TB), TB, 0, stream>>>(src, dst, EL, ER, Menc, E, H);
        gat_edge_sum<<<gblk(E*H, TB), TB, 0, stream>>>(src, dst, EL, ER, Menc, S, E, H);
        gat_edge_coef<<<gblk(E*H, TB), TB, 0, stream>>>(src, dst, EL, ER, Menc, S, AL, E, H);
        gat_edge_aggr<<<gblk(E*H*(Fo>>2), TB), TB, 0, stream>>>(src, dst, AL, FS2, HA, E, H, Fo);
        gat_finalize<<<gblk(N*HF, TB), TB, 0, stream>>>(HA, RES2, b2, out, N*HF, HF, 0);
    }
}